// MeshLaplacianLoss_8117488189441
// MI455X (gfx1250) — compile-verified
//
#include <hip/hip_runtime.h>

#define BB   8
#define NN   262144
#define FF   524288
#define BN3  (BB * NN * 3)   // 6,291,456 elements
#define Q    (BB * 3)        // 24 channels per vertex (vertex-major layout)

typedef __attribute__((ext_vector_type(2))) float v2f;
typedef __attribute__((ext_vector_type(8))) float v8f;

// ---------------- zero workspace + output ----------------
__global__ void zero_ws_kernel(float* __restrict__ acc, unsigned* __restrict__ deg,
                               float* __restrict__ out) {
    long long idx = (long long)blockIdx.x * blockDim.x + threadIdx.x;
    if (idx < BN3)                acc[idx] = 0.0f;
    else if (idx < BN3 + NN)      deg[idx - BN3] = 0u;
    else if (idx == BN3 + NN)     out[0] = 0.0f;
}

// ---------------- d = vert1 - vert2, transposed to [N][B*3] ----------------
__global__ void diff_kernel(const float* __restrict__ v1, const float* __restrict__ v2,
                            float* __restrict__ dt) {
    long long idx = (long long)blockIdx.x * blockDim.x + threadIdx.x; // b*N*3 + n*3 + c
    if (idx >= BN3) return;
    int c = (int)(idx % 3);
    long long t = idx / 3;
    int n = (int)(t % NN);
    int b = (int)(t / NN);
    dt[(long long)n * Q + b * 3 + c] = v1[idx] - v2[idx];
}

// ---------------- degree: each face adds 2 to each of its vertices ----------------
__global__ void deg_kernel(const long long* __restrict__ face, unsigned* __restrict__ deg) {
    int f = blockIdx.x * blockDim.x + threadIdx.x;
    if (f >= FF) return;
    int i = (int)face[3 * f + 0];
    int j = (int)face[3 * f + 1];
    int k = (int)face[3 * f + 2];
    atomicAdd(&deg[i], 2u);
    atomicAdd(&deg[j], 2u);
    atomicAdd(&deg[k], 2u);
}

// ---------------- scatter: acc[i]+=d[j]+d[k]; acc[j]+=d[i]+d[k]; acc[k]+=d[i]+d[j]
// 24 consecutive threads = 24 channels of one face -> 96B coalesced gathers/atomics.
__global__ void scatter_kernel(const long long* __restrict__ face,
                               const float* __restrict__ dt,
                               float* __restrict__ acc) {
    long long t = (long long)blockIdx.x * blockDim.x + threadIdx.x;
    int f = (int)(t / Q);
    int q = (int)(t % Q);
    if (f >= FF) return;
    if (f + 256 < FF) __builtin_prefetch(&face[3 * (f + 256)], 0, 1); // global_prefetch_b8
    long long i = face[3 * f + 0];
    long long j = face[3 * f + 1];
    long long k = face[3 * f + 2];
    float di = dt[i * Q + q];
    float dj = dt[j * Q + q];
    float dk = dt[k * Q + q];
    atomicAdd(&acc[i * Q + q], dj + dk);
    atomicAdd(&acc[j * Q + q], di + dk);
    atomicAdd(&acc[k * Q + q], di + dj);
}

// ---------------- finalize: mean |acc/deg - d| with WMMA wave reduction ----------------
__global__ void finalize_kernel(const float* __restrict__ acc, const float* __restrict__ dt,
                                const unsigned* __restrict__ deg, float* __restrict__ out) {
    const float inv_total = 1.0f / (float)BN3;
    const long long total = (long long)NN * Q;            // 6,291,456
    const long long stride = (long long)gridDim.x * blockDim.x;
    float p = 0.0f;
    // 1024 blocks * 256 threads -> exactly 24 iterations per thread, no divergence
    for (long long idx = (long long)blockIdx.x * blockDim.x + threadIdx.x;
         idx < total; idx += stride) {
        int n = (int)(idx / Q);
        float degf = fmaxf((float)deg[n], 1.0f);
        float v = acc[idx] / degf - dt[idx];
        p += fabsf(v);
    }

    // Wave-level sum via V_WMMA_F32_16X16X4_F32 with a ones B-matrix:
    // A: VGPR0 = p (K0 for lanes 0-15, K2 for lanes 16-31), VGPR1 = 0.
    // D[m][n] = p_m + p_{m+16}; sum the 8 D VGPRs then fold with shfl_xor(16).
    v2f a;  a[0] = p;    a[1] = 0.0f;
    v2f bo; bo[0] = 1.0f; bo[1] = 1.0f;
    v8f cz = {};
    v8f dm = __builtin_amdgcn_wmma_f32_16x16x4_f32(false, a, false, bo,
                                                   (short)0, cz, false, false);
    float tsum = dm[0] + dm[1] + dm[2] + dm[3] + dm[4] + dm[5] + dm[6] + dm[7];
    tsum += __shfl_xor(tsum, 16, 32);                     // full 32-lane wave sum

    __shared__ float sh[8];
    int lane = threadIdx.x & 31;
    int wid  = threadIdx.x >> 5;
    if (lane == 0) sh[wid] = tsum;
    __syncthreads();
    if (threadIdx.x == 0) {
        float bsum = 0.0f;
        int nw = blockDim.x >> 5;
        for (int w = 0; w < nw; ++w) bsum += sh[w];
        atomicAdd(out, bsum * inv_total);
    }
}

extern "C" void kernel_launch(void* const* d_in, const int* in_sizes, int n_in,
                              void* d_out, int out_size, void* d_ws, size_t ws_size,
                              hipStream_t stream) {
    const float*     v1   = (const float*)d_in[0];
    const float*     v2   = (const float*)d_in[1];
    const long long* face = (const long long*)d_in[2];   // int64 faces
    float* out = (float*)d_out;

    float*    dt  = (float*)d_ws;            // [N][24]  25.2 MB
    float*    acc = dt + BN3;                // [N][24]  25.2 MB
    unsigned* deg = (unsigned*)(acc + BN3);  // [N]       1.0 MB

    {
        long long tot = (long long)BN3 + NN + 1;
        zero_ws_kernel<<<(int)((tot + 255) / 256), 256, 0, stream>>>(acc, deg, out);
    }
    diff_kernel<<<(BN3 + 255) / 256, 256, 0, stream>>>(v1, v2, dt);
    deg_kernel<<<(FF + 255) / 256, 256, 0, stream>>>(face, deg);
    {
        long long tot = (long long)FF * Q;   // 12,582,912 -> 49,152 blocks exactly
        scatter_kernel<<<(int)(tot / 256), 256, 0, stream>>>(face, dt, acc);
    }
    finalize_kernel<<<1024, 256, 0, stream>>>(acc, dt, deg, out);
}